// StreetMoverDistance_90177133347549
// MI455X (gfx1250) — compile-verified
//
#include <hip/hip_runtime.h>
#include <hip/hip_bf16.h>
#include <math.h>

typedef __attribute__((ext_vector_type(2))) float v2f;
typedef __attribute__((ext_vector_type(8))) float v8f;

#define EPS_SK      1.0f
#define INV_EPS_SK  1.0f
#define MAX_ITER_SK 10
#define THRESH_SK   0.1f

// ---------------------------------------------------------------------------
// Edge index decode: linear tril(-1) index e -> (i, j), row-major nested-loop
// order: e = i*(i-1)/2 + j, j < i.
// ---------------------------------------------------------------------------
__device__ inline void decode_edge(long long e, int& i, long long& j) {
    int ii = (int)((1.0 + sqrt(1.0 + 8.0 * (double)e)) * 0.5);
    while ((long long)ii * (ii - 1) / 2 > e) --ii;
    while ((long long)(ii + 1) * ii / 2 <= e) ++ii;
    i = ii;
    j = e - (long long)ii * (ii - 1) / 2;
}

// ---------------------------------------------------------------------------
// Phase 1: point cloud (edge lengths -> prefix sum -> searchsorted + lerp)
// ---------------------------------------------------------------------------
__global__ void edge_len_kernel(const int* __restrict__ A, const float* __restrict__ nodes,
                                float* __restrict__ cum, int N, long long E) {
    long long e = (long long)blockIdx.x * 256 + threadIdx.x;
    if (e >= E) return;
    int i; long long j;
    decode_edge(e, i, j);
    float len = 0.0f;
    if (A[(long long)i * N + j] == 1) {
        float dx = nodes[2 * i]     - nodes[2 * j];
        float dy = nodes[2 * i + 1] - nodes[2 * j + 1];
        len = sqrtf(dx * dx + dy * dy);
    }
    cum[e] = len;
}

// Per-2048-element-chunk inclusive scan; emits chunk totals.
__global__ void scan_chunk_kernel(float* __restrict__ data, float* __restrict__ chunkSums,
                                  long long E) {
    __shared__ float ssum[256];
    long long base = (long long)blockIdx.x * 2048 + (long long)threadIdx.x * 8;
    float vals[8];
    float run = 0.0f;
#pragma unroll
    for (int k = 0; k < 8; ++k) {
        long long idx = base + k;
        float v = (idx < E) ? data[idx] : 0.0f;
        run += v;
        vals[k] = run;
    }
    ssum[threadIdx.x] = run;
    __syncthreads();
    for (int off = 1; off < 256; off <<= 1) {
        float t = (threadIdx.x >= off) ? ssum[threadIdx.x - off] : 0.0f;
        __syncthreads();
        ssum[threadIdx.x] += t;
        __syncthreads();
    }
    float excl = (threadIdx.x > 0) ? ssum[threadIdx.x - 1] : 0.0f;
#pragma unroll
    for (int k = 0; k < 8; ++k) {
        long long idx = base + k;
        if (idx < E) data[idx] = vals[k] + excl;
    }
    if (threadIdx.x == 255) chunkSums[blockIdx.x] = ssum[255];
}

// Single-block inclusive scan of chunk totals (n <= 1024).
__global__ void scan_sums_kernel(float* __restrict__ chunkSums, int n) {
    __shared__ float s[1024];
    int t = threadIdx.x;
    s[t] = (t < n) ? chunkSums[t] : 0.0f;
    __syncthreads();
    for (int off = 1; off < 1024; off <<= 1) {
        float tv = (t >= off) ? s[t - off] : 0.0f;
        __syncthreads();
        s[t] += tv;
        __syncthreads();
    }
    if (t < n) chunkSums[t] = s[t];
}

__global__ void add_off_kernel(float* __restrict__ data, const float* __restrict__ chunkSums,
                               long long E) {
    long long idx = (long long)blockIdx.x * 256 + threadIdx.x;
    if (idx >= E) return;
    int c = (int)(idx >> 11);
    if (c > 0) data[idx] += chunkSums[c - 1];
}

__global__ void sample_kernel(const int* __restrict__ A, const float* __restrict__ nodes,
                              const float* __restrict__ cum, const int* __restrict__ npts_p,
                              float* __restrict__ pc, int N, long long E, int P) {
    int k = blockIdx.x * 256 + threadIdx.x;
    if (k >= P) return;
    float total = cum[E - 1];
    if (!(total > 0.0f)) { pc[2 * k] = 0.0f; pc[2 * k + 1] = 0.0f; return; }
    int np = *npts_p;
    float step = total / ((float)(np - 1) + 0.01f);
    float s = (float)k * step;
    // searchsorted(cum, s, side='right') == upper_bound
    long long lo = 0, hi = E;
    while (lo < hi) {
        long long mid = (lo + hi) >> 1;
        if (cum[mid] <= s) lo = mid + 1; else hi = mid;
    }
    long long e = lo; if (e > E - 1) e = E - 1;
    int i; long long j;
    decode_edge(e, i, j);
    float segx = nodes[2 * i]     - nodes[2 * j];
    float segy = nodes[2 * i + 1] - nodes[2 * j + 1];
    float ln = 0.0f;
    if (A[(long long)i * N + j] == 1) ln = sqrtf(segx * segx + segy * segy);
    float start = cum[e] - ln;
    float t = (s - start) / fmaxf(ln, 1e-12f);
    t = fminf(fmaxf(t, 0.0f), 1.0f);
    pc[2 * k]     = nodes[2 * j]     + t * segx;
    pc[2 * k + 1] = nodes[2 * j + 1] + t * segy;
}

// ---------------------------------------------------------------------------
// Phase 2: Sinkhorn setup + gated iterations (LDS-resident operands)
// ---------------------------------------------------------------------------
__global__ void prep_kernel(const float* __restrict__ x, const float* __restrict__ y,
                            float* __restrict__ u, float* __restrict__ v,
                            float* __restrict__ nx, float* __restrict__ ny,
                            float* __restrict__ err, int* __restrict__ done,
                            float* __restrict__ cost, int P) {
    int i = blockIdx.x * 256 + threadIdx.x;
    if (i < P) {
        float a = x[2 * i], b = x[2 * i + 1];
        nx[i] = a * a + b * b;
        a = y[2 * i]; b = y[2 * i + 1];
        ny[i] = a * a + b * b;
        u[i] = 0.0f;
        v[i] = 0.0f;
    }
    if (blockIdx.x == 0 && threadIdx.x == 0) { *err = 0.0f; *done = 0; *cost = 0.0f; }
}

// Row update: u_new[i] = EPS*(log_mu - LSE_j((-C_ij + u_i + v_j)/EPS)) + u_i
// Wave-per-row, y-points + v staged in LDS, two-pass (max, sum-exp) LSE.
__global__ void sink_row_kernel(const float* __restrict__ x, const float* __restrict__ y,
                                const float* __restrict__ u, const float* __restrict__ v,
                                float* __restrict__ u2, float* __restrict__ err,
                                const int* __restrict__ done, int P) {
    if (*done) return;
    extern __shared__ float smem[];
    float* sy = smem;          // 2P floats
    float* sv = smem + 2 * P;  // P floats
    int tid = threadIdx.y * 32 + threadIdx.x;
    for (int t = tid; t < 2 * P; t += 256) sy[t] = y[t];
    for (int t = tid; t < P; t += 256) sv[t] = v[t];
    __syncthreads();

    int lane = threadIdx.x;
    int i = blockIdx.x * 8 + threadIdx.y;
    float xi0 = x[2 * i], xi1 = x[2 * i + 1], ui = u[i];

    float m = -__builtin_inff();
    for (int j = lane; j < P; j += 32) {
        float dx = xi0 - sy[2 * j], dy = xi1 - sy[2 * j + 1];
        float mij = (ui + sv[j] - (dx * dx + dy * dy)) * INV_EPS_SK;
        m = fmaxf(m, mij);
    }
    for (int off = 16; off; off >>= 1) m = fmaxf(m, __shfl_xor(m, off));

    float s = 0.0f;
    for (int j = lane; j < P; j += 32) {
        float dx = xi0 - sy[2 * j], dy = xi1 - sy[2 * j + 1];
        float mij = (ui + sv[j] - (dx * dx + dy * dy)) * INV_EPS_SK;
        s += expf(mij - m);
    }
    for (int off = 16; off; off >>= 1) s += __shfl_xor(s, off);

    if (lane == 0) {
        float lse = m + logf(s);
        float log_mu = logf(1.0f / (float)P + 1e-8f);
        float un = EPS_SK * (log_mu - lse) + ui;
        u2[i] = un;
        atomicAdd(err, fabsf(un - ui));
    }
}

// Column update (symmetric: reduce over i using x-points and u_new).
__global__ void sink_col_kernel(const float* __restrict__ x, const float* __restrict__ y,
                                const float* __restrict__ u2, const float* __restrict__ v,
                                float* __restrict__ v2, const int* __restrict__ done, int P) {
    if (*done) return;
    extern __shared__ float smem[];
    float* sx = smem;          // 2P floats
    float* su = smem + 2 * P;  // P floats
    int tid = threadIdx.y * 32 + threadIdx.x;
    for (int t = tid; t < 2 * P; t += 256) sx[t] = x[t];
    for (int t = tid; t < P; t += 256) su[t] = u2[t];
    __syncthreads();

    int lane = threadIdx.x;
    int j = blockIdx.x * 8 + threadIdx.y;
    float yj0 = y[2 * j], yj1 = y[2 * j + 1], vj = v[j];

    float m = -__builtin_inff();
    for (int i = lane; i < P; i += 32) {
        float dx = sx[2 * i] - yj0, dy = sx[2 * i + 1] - yj1;
        float mij = (su[i] + vj - (dx * dx + dy * dy)) * INV_EPS_SK;
        m = fmaxf(m, mij);
    }
    for (int off = 16; off; off >>= 1) m = fmaxf(m, __shfl_xor(m, off));

    float s = 0.0f;
    for (int i = lane; i < P; i += 32) {
        float dx = sx[2 * i] - yj0, dy = sx[2 * i + 1] - yj1;
        float mij = (su[i] + vj - (dx * dx + dy * dy)) * INV_EPS_SK;
        s += expf(mij - m);
    }
    for (int off = 16; off; off >>= 1) s += __shfl_xor(s, off);

    if (lane == 0) {
        float lse = m + logf(s);
        float log_nu = logf(1.0f / (float)P + 1e-8f);
        v2[j] = EPS_SK * (log_nu - lse) + vj;
    }
}

__global__ void copy_uv_kernel(float* __restrict__ u, const float* __restrict__ u2,
                               float* __restrict__ v, const float* __restrict__ v2,
                               const int* __restrict__ done, int P) {
    if (*done) return;
    int i = blockIdx.x * 256 + threadIdx.x;
    if (i < P) { u[i] = u2[i]; v[i] = v2[i]; }
}

__global__ void done_kernel(float* __restrict__ err, int* __restrict__ done) {
    if (*done) return;
    if (*err < THRESH_SK) *done = 1;
    *err = 0.0f;
}

// ---------------------------------------------------------------------------
// Phase 3: pi = exp(M(u,v)), C, cost = sum(pi*C).
// One wave handles a 16x64 output strip: 1 A-fragment, 4x V_WMMA_F32_16X16X4_F32
// (K=2 padded to 4), row scalars (nx,u) hoisted across the 4 column tiles.
// A 16x4 f32 layout: VGPR0 = K0 (lanes 0-15, M=lane) / K2 (lanes 16-31),
//                    VGPR1 = K1 / K3.  B 4x16 mirrors with N across lanes.
// D: lane<16 -> (M=r, N=lane); lane>=16 -> (M=8+r, N=lane-16).
// ---------------------------------------------------------------------------
__global__ void sink_final_kernel(const float* __restrict__ x, const float* __restrict__ y,
                                  const float* __restrict__ u, const float* __restrict__ v,
                                  const float* __restrict__ nx, const float* __restrict__ ny,
                                  float* __restrict__ pi, float* __restrict__ Cm,
                                  float* __restrict__ cost, int P) {
    const int lane = threadIdx.x;                 // 0..31
    const int tI = blockIdx.x;                    // row tile
    const int tJ0 = (blockIdx.y * blockDim.y + threadIdx.y) * 4;  // first col tile (wave-uniform)
    if (tJ0 * 16 >= P) return;
    const int r16 = lane & 15;
    const bool hi = lane >= 16;
    const int mbase = hi ? 8 : 0;

    v2f a = {0.0f, 0.0f};
    if (!hi) {  // K=2,3 rows are zero padding
        int i = tI * 16 + r16;
        a.x = x[2 * i];
        a.y = x[2 * i + 1];
    }

    // Row scalars reused across all 4 column tiles.
    float nxr[8], ur[8];
#pragma unroll
    for (int r = 0; r < 8; ++r) {
        int i = tI * 16 + mbase + r;
        nxr[r] = nx[i];
        ur[r] = u[i];
    }

    float acc = 0.0f;
#pragma unroll
    for (int t = 0; t < 4; ++t) {
        const int tJ = tJ0 + t;
        v2f b = {0.0f, 0.0f};
        if (!hi) {
            int j = tJ * 16 + r16;
            b.x = y[2 * j];
            b.y = y[2 * j + 1];
        }
        v8f c = {};
        // d[r] = dot(x_{tI*16 + mbase + r}, y_{tJ*16 + (lane&15)})
        v8f d = __builtin_amdgcn_wmma_f32_16x16x4_f32(
            /*neg_a=*/false, a, /*neg_b=*/false, b,
            /*c_mod=*/(short)0, c, /*reuse_a=*/false, /*reuse_b=*/false);

        const int j = tJ * 16 + r16;
        const float vj = v[j];
        const float nyj = ny[j];
#pragma unroll
        for (int r = 0; r < 8; ++r) {
            int i = tI * 16 + mbase + r;
            float Cij = nxr[r] + nyj - 2.0f * d[r];
            float pij = expf((vj + ur[r] - Cij) * INV_EPS_SK);
            size_t off = (size_t)i * (size_t)P + (size_t)j;
            Cm[off] = Cij;
            pi[off] = pij;
            acc += pij * Cij;
        }
    }
    for (int off = 16; off; off >>= 1) acc += __shfl_xor(acc, off);
    if (lane == 0) atomicAdd(cost, acc);
}

// ---------------------------------------------------------------------------
// Host launcher
// ---------------------------------------------------------------------------
extern "C" void kernel_launch(void* const* d_in, const int* in_sizes, int n_in,
                              void* d_out, int out_size, void* d_ws, size_t ws_size,
                              hipStream_t stream) {
    // N from adjacency size, P from out_size = 2P^2 + 4P + 1
    int N = (int)(sqrt((double)in_sizes[0]) + 0.5);
    while ((long long)N * N > in_sizes[0]) --N;
    while ((long long)(N + 1) * (N + 1) <= in_sizes[0]) ++N;
    long long os = (long long)out_size;
    int P = (int)((sqrt(8.0 * (double)os + 8.0) - 4.0) / 4.0 + 0.5);
    while (2LL * P * P + 4LL * P + 1 > os) --P;
    while (2LL * (P + 1) * (P + 1) + 4LL * (P + 1) + 1 <= os) ++P;

    const long long E = (long long)N * (N - 1) / 2;
    const int nchunks = (int)((E + 2047) / 2048);   // <= 1024 for N <= 2048

    const int*   A0 = (const int*)d_in[0];
    const float* n0 = (const float*)d_in[1];
    const int*   A1 = (const int*)d_in[2];
    const float* n1 = (const float*)d_in[3];
    const int*   np = (const int*)d_in[4];

    float* out  = (float*)d_out;
    float* ypc  = out;                       // [P,2]
    float* opc  = out + 2 * (size_t)P;       // [P,2]
    float* cost = out + 4 * (size_t)P;       // [1]
    float* pi   = cost + 1;                  // [P,P]
    float* Cm   = pi + (size_t)P * P;        // [P,P]
    float* cumBuf = pi;                      // scratch: overwritten by final kernel

    float* wsf   = (float*)d_ws;
    float* u     = wsf;
    float* v     = wsf + (size_t)P;
    float* u2    = wsf + 2 * (size_t)P;
    float* v2    = wsf + 3 * (size_t)P;
    float* nx    = wsf + 4 * (size_t)P;
    float* ny    = wsf + 5 * (size_t)P;
    float* err   = wsf + 6 * (size_t)P;
    int*   done  = (int*)(wsf + 6 * (size_t)P + 1);
    float* csums = wsf + 6 * (size_t)P + 8;

    const int nblkE = (int)((E + 255) / 256);
    const int nblkP = (P + 255) / 256;

    // --- point clouds (graph 0 -> ypc, graph 1 -> opc) ---
    edge_len_kernel<<<nblkE, 256, 0, stream>>>(A0, n0, cumBuf, N, E);
    scan_chunk_kernel<<<nchunks, 256, 0, stream>>>(cumBuf, csums, E);
    scan_sums_kernel<<<1, 1024, 0, stream>>>(csums, nchunks);
    add_off_kernel<<<nblkE, 256, 0, stream>>>(cumBuf, csums, E);
    sample_kernel<<<nblkP, 256, 0, stream>>>(A0, n0, cumBuf, np, ypc, N, E, P);

    edge_len_kernel<<<nblkE, 256, 0, stream>>>(A1, n1, cumBuf, N, E);
    scan_chunk_kernel<<<nchunks, 256, 0, stream>>>(cumBuf, csums, E);
    scan_sums_kernel<<<1, 1024, 0, stream>>>(csums, nchunks);
    add_off_kernel<<<nblkE, 256, 0, stream>>>(cumBuf, csums, E);
    sample_kernel<<<nblkP, 256, 0, stream>>>(A1, n1, cumBuf, np, opc, N, E, P);

    // --- sinkhorn ---
    prep_kernel<<<nblkP, 256, 0, stream>>>(ypc, opc, u, v, nx, ny, err, done, cost, P);

    const size_t smemBytes = (size_t)3 * P * sizeof(float);  // 48 KB at P=4096
    dim3 wblk(32, 8);
    for (int it = 0; it < MAX_ITER_SK; ++it) {
        sink_row_kernel<<<P / 8, wblk, smemBytes, stream>>>(ypc, opc, u, v, u2, err, done, P);
        sink_col_kernel<<<P / 8, wblk, smemBytes, stream>>>(ypc, opc, u2, v, v2, done, P);
        copy_uv_kernel<<<nblkP, 256, 0, stream>>>(u, u2, v, v2, done, P);
        done_kernel<<<1, 1, 0, stream>>>(err, done);
    }

    // --- outputs: pi, C, cost (4x WMMA f32 16x16x4 per wave for x.y^T) ---
    dim3 fblk(32, 8);
    int tilesJ = P / 16;                 // column tiles
    int stripsJ = (tilesJ + 3) / 4;      // 4 col tiles per wave
    dim3 fgrd(P / 16, (stripsJ + 7) / 8);
    sink_final_kernel<<<fgrd, fblk, 0, stream>>>(ypc, opc, u, v, nx, ny, pi, Cm, cost, P);
}